// CosVQ_EMA_1657857376704
// MI455X (gfx1250) — compile-verified
//
#include <hip/hip_runtime.h>

// ---------------------------------------------------------------------------
// CosVQ-EMA for MI455X (gfx1250, wave32, WMMA).
// scores = l2norm(z) @ l2norm(emb)^T : 34.4 GFLOP f32 GEMM via
// V_WMMA_F32_16X16X4_F32. Operands stored transposed AND pair-interleaved
// ([d/2][n][2]) so every WMMA fragment is a single b64 load. Each wave runs
// two independent accumulator chains (2 column tiles) for WMMA latency ILP.
// Softmax needs no max-subtraction: cosine scores in [-1,1], exp(s/T)<=e^10.
// ---------------------------------------------------------------------------

#define N_TOK   16384
#define DIMV    128
#define KCB     8192
#define INV_T   10.0f      // 1/TEMP
#define DECAY_A 0.8f
#define DECAY_B 0.2f

typedef __attribute__((ext_vector_type(2))) float v2f;
typedef __attribute__((ext_vector_type(8))) float v8f;

// ---- workspace layout (float offsets) ----
#define WS_ZNP   0                          // [64][N_TOK][2] paired zn^T
#define WS_ENP   (N_TOK * DIMV)             // [64][KCB][2]   paired en^T
#define WS_IDX   (WS_ENP + KCB * DIMV)      // N ints
#define WS_CNT   (WS_IDX + N_TOK)           // K counts
#define WS_SUMS  (WS_CNT + KCB)             // K*D per-code z sums
#define WS_PSUM  (WS_SUMS + KCB * DIMV)     // K column prob sums
#define WS_SCAL  (WS_PSUM + KCB)            // [0]=sse [1]=pplSum [2]=entSum
#define ZERO_BEG WS_CNT
#define ZERO_LEN (WS_SCAL + 4 - WS_CNT)

// ---- output layout (float offsets, concat of tuple) ----
#define OUT_ZQ     0
#define OUT_COMMIT (N_TOK * DIMV)
#define OUT_PPL    (OUT_COMMIT + 1)
#define OUT_ENT    (OUT_COMMIT + 2)
#define OUT_CS     (OUT_COMMIT + 3)
#define OUT_EMA    (OUT_CS + KCB)
#define OUT_EMB    (OUT_EMA + KCB * DIMV)

// ---------------------------------------------------------------------------
__global__ void zero_kernel(float* __restrict__ p, int n) {
    for (int i = blockIdx.x * blockDim.x + threadIdx.x; i < n;
         i += gridDim.x * blockDim.x)
        p[i] = 0.0f;
}

// One wave per vector; writes transposed + pair-interleaved normalized output:
// dst[(pair * stride + col) * 2 + {0,1}] = v[2*pair], v[2*pair+1]
__global__ __launch_bounds__(256) void normalize_kernel(
    const float* __restrict__ z, const float* __restrict__ emb,
    float* __restrict__ znP, float* __restrict__ enP) {
    const int lane = threadIdx.x & 31;
    const int wave = threadIdx.x >> 5;
    const int gid  = blockIdx.x * 8 + wave;

    const float* src;
    float* dst;
    int stride, col;
    if (gid < N_TOK) {
        src = z + (size_t)gid * DIMV; dst = znP; stride = N_TOK; col = gid;
    } else {
        int e = gid - N_TOK;
        if (e >= KCB) return;
        src = emb + (size_t)e * DIMV; dst = enP; stride = KCB; col = e;
    }
    float4 x = *(const float4*)(src + lane * 4);
    float ss = x.x * x.x + x.y * x.y + x.z * x.z + x.w * x.w;
#pragma unroll
    for (int m = 1; m < 32; m <<= 1) ss += __shfl_xor(ss, m, 32);
    float den = fmaxf(sqrtf(ss), 1e-12f);   // matches F.normalize eps clamp
    int p0 = lane * 2;                       // pair indices p0, p0+1
    float2 s0 = make_float2(x.x / den, x.y / den);
    float2 s1 = make_float2(x.z / den, x.w / den);
    *(float2*)&dst[((size_t)(p0 + 0) * stride + col) * 2] = s0;
    *(float2*)&dst[((size_t)(p0 + 1) * stride + col) * 2] = s1;
}

// ---------------------------------------------------------------------------
// Main WMMA kernel. Block = 8 waves = 16 rows of zn. Each wave owns TWO 16x16
// score tiles per iteration (independent accumulator chains), 32 chained
// V_WMMA_F32_16X16X4_F32 over D=128 per tile.
// Phase 1: per-row argmax + sum(exp(s/T)).  Phase 2: recompute tiles,
// accumulate per-column softmax prob sums (for P_avg / entropy).
// ---------------------------------------------------------------------------
__global__ __launch_bounds__(256) void scores_kernel(
    const float* __restrict__ znP, const float* __restrict__ enP,
    int* __restrict__ idx_out, float* __restrict__ P_sum) {
    const int lane    = threadIdx.x & 31;
    const int wave    = threadIdx.x >> 5;
    const int nlo     = lane & 15;
    const int laneHi  = lane >> 4;          // 0 | 1 (lane half)
    const int rowBase = blockIdx.x * 16;

    // Preload A fragments once (reused for all column tiles, both phases).
    // ISA 32-bit A 16x4 layout: VGPR0 = K {0|2}, VGPR1 = K {1|3} by lane half.
    // pair index = 2*s + laneHi  ->  elements d0=4s+2h, d0+1.
    v2f a[32];
#pragma unroll
    for (int s = 0; s < 32; ++s)
        a[s] = *(const v2f*)&znP[((size_t)(2 * s + laneHi) * N_TOK +
                                  rowBase + nlo) * 2];

    float expAcc[8], maxV[8];
    int   maxC[8];
#pragma unroll
    for (int j = 0; j < 8; ++j) { expAcc[j] = 0.0f; maxV[j] = -1e30f; maxC[j] = 0; }

    const int NIT = KCB / 256;              // 8 waves * 32 cols each

    // ---------------- Phase 1: argmax + row sum of exp(s/T) ----------------
    for (int it = 0; it < NIT; ++it) {
        const int colBase = it * 256 + wave * 32;
        if (it + 1 < NIT)  // warm L0 for next iteration's B slice
            __builtin_prefetch(&enP[((size_t)laneHi * KCB +
                                     (it + 1) * 256 + wave * 32 + nlo) * 2], 0, 1);
        v8f c0 = {}, c1 = {};
#pragma unroll
        for (int s = 0; s < 32; ++s) {
            size_t pr = ((size_t)(2 * s + laneHi) * KCB + colBase + nlo) * 2;
            v2f b0 = *(const v2f*)&enP[pr];
            v2f b1 = *(const v2f*)&enP[pr + 32];    // +16 cols * 2
            c0 = __builtin_amdgcn_wmma_f32_16x16x4_f32(
                     false, a[s], false, b0, (short)0, c0, false, false);
            c1 = __builtin_amdgcn_wmma_f32_16x16x4_f32(
                     false, a[s], false, b1, (short)0, c1, false, false);
        }
        const int col0 = colBase + nlo;     // C: lane holds col, c[j] = row j(+8)
        const int col1 = col0 + 16;
#pragma unroll
        for (int j = 0; j < 8; ++j) {
            float v0 = c0[j], v1 = c1[j];
            expAcc[j] += __expf(v0 * INV_T) + __expf(v1 * INV_T);
            if (v0 > maxV[j]) { maxV[j] = v0; maxC[j] = col0; }
            if (v1 > maxV[j]) { maxV[j] = v1; maxC[j] = col1; }
        }
    }

    // reduce across the 16 lanes of each half-wave (per-row results)
#pragma unroll
    for (int j = 0; j < 8; ++j) {
#pragma unroll
        for (int m = 1; m < 16; m <<= 1) {
            float oS = __shfl_xor(expAcc[j], m, 32);
            float oV = __shfl_xor(maxV[j],  m, 32);
            int   oC = __shfl_xor(maxC[j],  m, 32);
            expAcc[j] += oS;
            if (oV > maxV[j] || (oV == maxV[j] && oC < maxC[j])) {
                maxV[j] = oV; maxC[j] = oC;
            }
        }
    }

    __shared__ float sSum[16][8];
    __shared__ float sMax[16][8];
    __shared__ int   sArg[16][8];
    __shared__ float sRecipL[16];
    if (nlo == 0) {                      // lanes 0 and 16 carry the results
#pragma unroll
        for (int j = 0; j < 8; ++j) {
            int row = j + laneHi * 8;
            sSum[row][wave] = expAcc[j];
            sMax[row][wave] = maxV[j];
            sArg[row][wave] = maxC[j];
        }
    }
    __syncthreads();
    if (threadIdx.x < 16) {
        int row = threadIdx.x;
        float S = 0.0f, mv = -1e30f;
        int mc = 0;
        for (int w = 0; w < 8; ++w) {
            S += sSum[row][w];
            float v = sMax[row][w]; int ci = sArg[row][w];
            if (v > mv || (v == mv && ci < mc)) { mv = v; mc = ci; }
        }
        idx_out[rowBase + row] = mc;
        sRecipL[row] = 1.0f / S;
    }
    __syncthreads();

    float rl[8];
#pragma unroll
    for (int j = 0; j < 8; ++j) rl[j] = sRecipL[j + laneHi * 8];

    // ---------------- Phase 2: column sums of softmax probs ----------------
    for (int it = 0; it < NIT; ++it) {
        const int colBase = it * 256 + wave * 32;
        v8f c0 = {}, c1 = {};
#pragma unroll
        for (int s = 0; s < 32; ++s) {
            size_t pr = ((size_t)(2 * s + laneHi) * KCB + colBase + nlo) * 2;
            v2f b0 = *(const v2f*)&enP[pr];
            v2f b1 = *(const v2f*)&enP[pr + 32];
            c0 = __builtin_amdgcn_wmma_f32_16x16x4_f32(
                     false, a[s], false, b0, (short)0, c0, false, false);
            c1 = __builtin_amdgcn_wmma_f32_16x16x4_f32(
                     false, a[s], false, b1, (short)0, c1, false, false);
        }
        float cs0 = 0.0f, cs1 = 0.0f;
#pragma unroll
        for (int j = 0; j < 8; ++j) {
            cs0 += __expf(c0[j] * INV_T) * rl[j];
            cs1 += __expf(c1[j] * INV_T) * rl[j];
        }
        cs0 += __shfl_xor(cs0, 16, 32);     // combine rows 0-7 / 8-15
        cs1 += __shfl_xor(cs1, 16, 32);
        if (laneHi == 0) {
            atomicAdd(&P_sum[colBase + nlo], cs0);
            atomicAdd(&P_sum[colBase + 16 + nlo], cs1);
        }
    }
}

// ---------------------------------------------------------------------------
// z_q gather (STE output), commit-loss SSE, code histogram and per-code sums.
__global__ __launch_bounds__(256) void gather_scatter_kernel(
    const float* __restrict__ z, const float* __restrict__ emb,
    const int* __restrict__ idx, float* __restrict__ out_zq,
    float* __restrict__ counts, float* __restrict__ sums,
    float* __restrict__ scal) {
    int t   = blockIdx.x * 256 + threadIdx.x;     // over N*D
    int row = t >> 7;
    int d   = t & 127;
    int k   = idx[row];
    float zf = z[t];
    float zq = emb[(size_t)k * DIMV + d];
    out_zq[t] = zf + (zq - zf);                   // z + sg(z_q - z), same rounding
    float diff = zf - zq;
    float sq = diff * diff;
#pragma unroll
    for (int m = 1; m < 32; m <<= 1) sq += __shfl_xor(sq, m, 32);
    if ((threadIdx.x & 31) == 0) atomicAdd(&scal[0], sq);
    atomicAdd(&sums[(size_t)k * DIMV + d], zf);
    if (d == 0) atomicAdd(&counts[k], 1.0f);
}

// Per-code: new_cluster_size output + perplexity / entropy partial sums.
__global__ __launch_bounds__(256) void codebook_kernel(
    const float* __restrict__ counts, const float* __restrict__ P_sum,
    const float* __restrict__ cs_in, float* __restrict__ out_cs,
    float* __restrict__ scal) {
    int k = blockIdx.x * 256 + threadIdx.x;
    float cnt    = counts[k];
    float e_mean = cnt * (1.0f / N_TOK);
    float t1     = e_mean * logf(e_mean + 1e-8f);       // perplexity term
    float P      = P_sum[k] * (1.0f / N_TOK) + 1e-8f;
    float t2     = P * logf(P);                          // entropy term
    out_cs[k]    = DECAY_A * cs_in[k] + DECAY_B * cnt;
#pragma unroll
    for (int m = 1; m < 32; m <<= 1) {
        t1 += __shfl_xor(t1, m, 32);
        t2 += __shfl_xor(t2, m, 32);
    }
    if ((threadIdx.x & 31) == 0) {
        atomicAdd(&scal[1], t1);
        atomicAdd(&scal[2], t2);
    }
}

// EMA embedding update + normalized new embedding.
__global__ __launch_bounds__(256) void ema_kernel(
    const float* __restrict__ ema_in, const float* __restrict__ cs_in,
    const float* __restrict__ counts, const float* __restrict__ sums,
    float* __restrict__ out_ema, float* __restrict__ out_emb) {
    int i = blockIdx.x * 256 + threadIdx.x;       // over K*D
    int k = i >> 7;
    float ncs  = DECAY_A * cs_in[k] + DECAY_B * counts[k];
    float nema = DECAY_A * ema_in[i] + DECAY_B * sums[i];
    out_ema[i] = nema;
    out_emb[i] = nema / fmaxf(ncs, 1e-5f);
}

__global__ void finalize_kernel(const float* __restrict__ scal,
                                float* __restrict__ out_scalars) {
    if (threadIdx.x == 0 && blockIdx.x == 0) {
        out_scalars[0] = 0.25f * scal[0] / (float)(N_TOK * DIMV); // commit_loss
        out_scalars[1] = expf(-scal[1]);                          // perplexity
        out_scalars[2] = -scal[2];                                // entropy_loss
    }
}

// ---------------------------------------------------------------------------
extern "C" void kernel_launch(void* const* d_in, const int* in_sizes, int n_in,
                              void* d_out, int out_size, void* d_ws, size_t ws_size,
                              hipStream_t stream) {
    const float* z      = (const float*)d_in[0];   // [4,4096,128]
    const float* emb    = (const float*)d_in[1];   // [8192,128]
    const float* cs_in  = (const float*)d_in[2];   // [8192]
    const float* ema_in = (const float*)d_in[3];   // [8192,128]
    float* out = (float*)d_out;
    float* ws  = (float*)d_ws;

    float* znP    = ws + WS_ZNP;
    float* enP    = ws + WS_ENP;
    int*   idx    = (int*)(ws + WS_IDX);
    float* counts = ws + WS_CNT;
    float* sums   = ws + WS_SUMS;
    float* P_sum  = ws + WS_PSUM;
    float* scal   = ws + WS_SCAL;

    zero_kernel<<<512, 256, 0, stream>>>(ws + ZERO_BEG, ZERO_LEN);
    normalize_kernel<<<(N_TOK + KCB) / 8, 256, 0, stream>>>(z, emb, znP, enP);
    scores_kernel<<<N_TOK / 16, 256, 0, stream>>>(znP, enP, idx, P_sum);
    gather_scatter_kernel<<<(N_TOK * DIMV) / 256, 256, 0, stream>>>(
        z, emb, idx, out + OUT_ZQ, counts, sums, scal);
    codebook_kernel<<<KCB / 256, 256, 0, stream>>>(counts, P_sum, cs_in,
                                                   out + OUT_CS, scal);
    ema_kernel<<<(KCB * DIMV) / 256, 256, 0, stream>>>(ema_in, cs_in, counts,
                                                       sums, out + OUT_EMA,
                                                       out + OUT_EMB);
    finalize_kernel<<<1, 32, 0, stream>>>(scal, out + OUT_COMMIT);
}